// GraphSAGE_42030549959219
// MI455X (gfx1250) — compile-verified
//
#include <hip/hip_runtime.h>
#include <hip/hip_bf16.h>
#include <math.h>

typedef __attribute__((ext_vector_type(2))) float v2f;
typedef __attribute__((ext_vector_type(8))) float v8f;

#define C 64            // all channel dims are 64
#define LDSW 66         // padded LDS row stride (keeps 8B alignment, spreads banks)

__device__ inline v8f wmma_f32(v2f a, v2f b, v8f c) {
    // D = A(16x4 f32) * B(4x16 f32) + C(16x16 f32)
    return __builtin_amdgcn_wmma_f32_16x16x4_f32(
        /*neg_a=*/false, a, /*neg_b=*/false, b,
        /*c_mod=*/(short)0, c, /*reuse_a=*/false, /*reuse_b=*/false);
}

// ---------------------------------------------------------------------------
// Zero a contiguous region of workspace (float4 granularity).
// ---------------------------------------------------------------------------
__global__ __launch_bounds__(256) void zero_kernel(float4* __restrict__ p, int n4) {
    int i = blockIdx.x * 256 + threadIdx.x;
    if (i < n4) p[i] = make_float4(0.f, 0.f, 0.f, 0.f);
}

// ---------------------------------------------------------------------------
// Edge scatter: one wave per edge. Lane handles channels (lane, lane+32).
// agg[dst] += feat[src]; optionally cnt[dst] += 1 (lane 0).
// ---------------------------------------------------------------------------
__global__ __launch_bounds__(256)
void scatter_kernel(const float* __restrict__ feat,
                    const int* __restrict__ src,
                    const int* __restrict__ dst,
                    float* __restrict__ agg,
                    float* __restrict__ cnt,
                    int n_edges, int do_cnt)
{
    int e = blockIdx.x * 8 + (threadIdx.x >> 5);
    if (e >= n_edges) return;
    int lane = threadIdx.x & 31;
    int s = src[e];
    int d = dst[e];
    float v0 = feat[(size_t)s * C + lane];
    float v1 = feat[(size_t)s * C + lane + 32];
    atomicAdd(&agg[(size_t)d * C + lane], v0);
    atomicAdd(&agg[(size_t)d * C + lane + 32], v1);
    if (do_cnt && lane == 0) atomicAdd(&cnt[d], 1.0f);
}

// ---------------------------------------------------------------------------
// Fused SAGE linear:
//   out[m][n] = act( (agg[m]/max(cnt[m],1)) @ Wl.T + bl + xin[m] @ Wr.T )
// One wave computes a 16x64 output tile via V_WMMA_F32_16X16X4_F32.
// Weights staged in LDS (row stride 66 to avoid bank conflicts).
// ---------------------------------------------------------------------------
__global__ __launch_bounds__(256)
void sage_linear_kernel(const float* __restrict__ agg,
                        const float* __restrict__ cnt,
                        const float* __restrict__ xin,
                        const float* __restrict__ Wl,   // [64][64] row-major (n,k)
                        const float* __restrict__ bl,   // [64]
                        const float* __restrict__ Wr,   // [64][64]
                        float* __restrict__ out,
                        int n_nodes, int relu)
{
    __shared__ float sWl[C * LDSW];
    __shared__ float sWr[C * LDSW];

    // Stage both weight matrices into LDS (coalesced global reads).
    for (int i = threadIdx.x; i < C * C; i += 256) {
        int n = i >> 6;
        int k = i & 63;
        sWl[n * LDSW + k] = Wl[i];
        sWr[n * LDSW + k] = Wr[i];
    }
    __syncthreads();

    const int lane = threadIdx.x & 31;
    const int l16  = lane & 15;     // N (for B/C/D) or M (for A) within tile
    const int half = lane >> 4;     // selects K pair (A/B) / M+8 (C/D)

    // Per-lane bias values: constant across the 8 rows of each C/D VGPR.
    float bias[4];
#pragma unroll
    for (int nt = 0; nt < 4; ++nt) bias[nt] = bl[nt * 16 + l16];

    const int n_tiles = n_nodes >> 4;           // 100000/16 = 6250 exact
    const int wave    = threadIdx.x >> 5;
    const int wstride = gridDim.x * 8;

    for (int tile = blockIdx.x * 8 + wave; tile < n_tiles; tile += wstride) {
        const int row = tile << 4;
        const int m   = row + l16;              // A-matrix row owned by this lane
        const float sc = 1.0f / fmaxf(cnt[m], 1.0f);

        v8f acc[4] = {v8f{}, v8f{}, v8f{}, v8f{}};

#pragma unroll
        for (int kt = 0; kt < 16; ++kt) {
            const int k = kt * 4 + half * 2;    // this lane's K, K+1 slots
            // A fragments: contiguous float2 along K
            v2f a1 = *(const v2f*)(agg + (size_t)m * C + k);
            a1 *= sc;                            // fused mean-divide
            v2f a2 = *(const v2f*)(xin + (size_t)m * C + k);
#pragma unroll
            for (int nt = 0; nt < 4; ++nt) {
                const int n = nt * 16 + l16;
                v2f bfl = *(const v2f*)(sWl + n * LDSW + k);  // Wl[n][k..k+1]
                v2f bfr = *(const v2f*)(sWr + n * LDSW + k);  // Wr[n][k..k+1]
                acc[nt] = wmma_f32(a1, bfl, acc[nt]);
                acc[nt] = wmma_f32(a2, bfr, acc[nt]);
            }
        }

        // Epilogue: bias, optional ReLU, store.
#pragma unroll
        for (int nt = 0; nt < 4; ++nt) {
            const int n = nt * 16 + l16;
#pragma unroll
            for (int i = 0; i < 8; ++i) {
                float v = acc[nt][i] + bias[nt];
                if (relu) v = fmaxf(v, 0.0f);
                out[(size_t)(row + half * 8 + i) * C + n] = v;
            }
        }
    }
}

// ---------------------------------------------------------------------------
// Row-wise L2 normalize (in place): one wave per row, shuffle reduction.
// ---------------------------------------------------------------------------
__global__ __launch_bounds__(256)
void l2norm_kernel(float* __restrict__ h, int n_nodes)
{
    int row = blockIdx.x * 8 + (threadIdx.x >> 5);
    if (row >= n_nodes) return;
    int lane = threadIdx.x & 31;
    float a = h[(size_t)row * C + lane];
    float b = h[(size_t)row * C + lane + 32];
    float s = a * a + b * b;
#pragma unroll
    for (int m = 16; m >= 1; m >>= 1) s += __shfl_xor(s, m, 32);
    float scale = 1.0f / fmaxf(sqrtf(s), 1e-12f);
    h[(size_t)row * C + lane]      = a * scale;
    h[(size_t)row * C + lane + 32] = b * scale;
}

// ---------------------------------------------------------------------------
// Driver
// ---------------------------------------------------------------------------
extern "C" void kernel_launch(void* const* d_in, const int* in_sizes, int n_in,
                              void* d_out, int out_size, void* d_ws, size_t ws_size,
                              hipStream_t stream)
{
    const float* x   = (const float*)d_in[0];
    const int*   ei  = (const int*)  d_in[1];
    const float* W1l = (const float*)d_in[2];
    const float* b1l = (const float*)d_in[3];
    const float* W1r = (const float*)d_in[4];
    const float* W2l = (const float*)d_in[5];
    const float* b2l = (const float*)d_in[6];
    const float* W2r = (const float*)d_in[7];
    float* out = (float*)d_out;

    const int N = in_sizes[0] / C;      // 100000
    const int E = in_sizes[1] / 2;      // 1600000
    const int* src = ei;
    const int* dst = ei + E;

    // Workspace layout (floats): agg1[N*64] | agg2[N*64] | cnt[N] | h1[N*64]
    float* ws   = (float*)d_ws;
    float* agg1 = ws;
    float* agg2 = agg1 + (size_t)N * C;
    float* cntv = agg2 + (size_t)N * C;
    float* h1   = cntv + N;

    // Zero agg1, agg2, cnt (contiguous region; 12.9M floats, /4 exact).
    size_t zf = (size_t)2 * N * C + N;
    int n4 = (int)(zf / 4);
    zero_kernel<<<(n4 + 255) / 256, 256, 0, stream>>>((float4*)ws, n4);

    const int sblocks = (E + 7) / 8;
    const int lblocks = 782;            // 6250 wave-tiles / 8 waves per block

    // Layer 1
    scatter_kernel<<<sblocks, 256, 0, stream>>>(x, src, dst, agg1, cntv, E, 1);
    sage_linear_kernel<<<lblocks, 256, 0, stream>>>(agg1, cntv, x, W1l, b1l, W1r,
                                                    h1, N, /*relu=*/1);
    // Layer 2
    scatter_kernel<<<sblocks, 256, 0, stream>>>(h1, src, dst, agg2, cntv, E, 0);
    sage_linear_kernel<<<lblocks, 256, 0, stream>>>(agg2, cntv, h1, W2l, b2l, W2r,
                                                    out, N, /*relu=*/0);
    // Row L2 normalize
    l2norm_kernel<<<(N + 7) / 8, 256, 0, stream>>>(out, N);
}